// InstrumentedModel_1099511628122
// MI455X (gfx1250) — compile-verified
//
#include <hip/hip_runtime.h>

typedef __attribute__((ext_vector_type(2))) float v2f;
typedef __attribute__((ext_vector_type(8))) float v8f;

#define C_DIM 128
#define L_NUM 2
#define NA_N 30000
#define NB_N 30000
#define E_N 150000
#define EPS_LN 1e-5f

// ---------------------------------------------------------------- utilities

__device__ __forceinline__ float wave_sum32(float v) {
  #pragma unroll
  for (int off = 16; off > 0; off >>= 1) v += __shfl_xor(v, off, 32);
  return v;
}

// CDNA5 async memory->LDS copy (16B per lane), tracked by ASYNCcnt.
// LDS address = low 32 bits of the generic shared pointer (AS(3) byte offset).
__device__ __forceinline__ void async_copy_b128(const float* gsrc, float* ldst) {
  unsigned lds_off = (unsigned)(size_t)ldst;
  unsigned long long gaddr = (unsigned long long)(size_t)gsrc;
  asm volatile("global_load_async_to_lds_b128 %0, %1, off"
               :: "v"(lds_off), "v"(gaddr) : "memory");
}
__device__ __forceinline__ void wait_async0() {
  asm volatile("s_wait_asynccnt 0x0" ::: "memory");
}

__global__ __launch_bounds__(256) void zero_kernel(float* __restrict__ p, int n) {
  int i = blockIdx.x * 256 + threadIdx.x;
  if (i < n) p[i] = 0.0f;
}

// nan->0, +inf->1, -inf->-1 (reference `enc`)
__global__ __launch_bounds__(256) void sanitize_in_kernel(const float* __restrict__ in,
                                                          float* __restrict__ out, int n) {
  int i = blockIdx.x * 256 + threadIdx.x;
  if (i >= n) return;
  float v = in[i];
  if (v != v) v = 0.0f;
  else if (__builtin_isinf(v)) v = (v > 0.0f) ? 1.0f : -1.0f;
  out[i] = v;
}

__global__ __launch_bounds__(256) void count_kernel(const int* __restrict__ dst,
                                                    float* __restrict__ cnt, int n) {
  int i = blockIdx.x * 256 + threadIdx.x;
  if (i < n) atomicAdd(&cnt[dst[i]], 1.0f);
}

// ------------------------------------------------- GEMM: Y = act(X@W + b), K=Nc=128
// One wave computes a 16x128 output strip with v_wmma_f32_16x16x4_f32.
// Block = 256 threads = 8 waves = 128 rows. W (128x128) staged in LDS via async copy.

__global__ __launch_bounds__(256) void gemm128_kernel(const float* __restrict__ X,
                                                      const float* __restrict__ W,
                                                      const float* __restrict__ bias,
                                                      float* __restrict__ Y,
                                                      int M, int do_relu) {
  __shared__ float sW[C_DIM * C_DIM];
  const int tid = threadIdx.x;
  for (int i = tid; i < C_DIM * C_DIM / 4; i += 256)
    async_copy_b128(W + 4 * i, sW + 4 * i);
  wait_async0();
  __syncthreads();

  const int wave = tid >> 5, lane = tid & 31;
  const int half = lane >> 4, l16 = lane & 15;
  const int row_base = blockIdx.x * 128 + wave * 16;
  const int rb = (row_base < M) ? row_base : (M - 16);   // wave-uniform clamp: EXEC all-ones
  const float* Xrow = X + (size_t)(rb + l16) * C_DIM + 2 * half;

  v8f acc[8] = {};
  v2f a_cur = *(const v2f*)(Xrow);
  for (int k0 = 0; k0 < C_DIM; k0 += 4) {
    const int kn = (k0 + 4) & (C_DIM - 1);               // wrap: always-valid prefetch
    v2f a_nxt = *(const v2f*)(Xrow + kn);
    const int ks = k0 + 2 * half;
    #pragma unroll
    for (int nt = 0; nt < 8; nt++) {
      const int n = nt * 16 + l16;
      v2f b;
      b[0] = sW[ks * C_DIM + n];
      b[1] = sW[(ks + 1) * C_DIM + n];
      acc[nt] = __builtin_amdgcn_wmma_f32_16x16x4_f32(false, a_cur, false, b,
                                                      (short)0, acc[nt], false, false);
    }
    a_cur = a_nxt;
  }
  if (row_base >= M) return;
  #pragma unroll
  for (int nt = 0; nt < 8; nt++) {
    const int n = nt * 16 + l16;
    const float bv = bias[n];
    #pragma unroll
    for (int r = 0; r < 8; r++) {
      float y = acc[nt][r] + bv;
      if (do_relu) y = fmaxf(y, 0.0f);
      Y[(size_t)(row_base + r + 8 * half) * C_DIM + n] = y;
    }
  }
}

// -------------------- update GEMM: Y = san10(relu([Xd | Acc/cnt] @ Wu + b)), K=256

__global__ __launch_bounds__(256) void update_gemm_kernel(const float* __restrict__ Xd,
                                                          const float* __restrict__ Acc,
                                                          const float* __restrict__ Cnt,
                                                          const float* __restrict__ Wu,
                                                          const float* __restrict__ bias,
                                                          float* __restrict__ Y, int M) {
  __shared__ float sW[C_DIM * C_DIM];                    // one 128-K chunk at a time
  const int tid = threadIdx.x;
  const int wave = tid >> 5, lane = tid & 31;
  const int half = lane >> 4, l16 = lane & 15;
  const int row_base = blockIdx.x * 128 + wave * 16;
  const int rb = (row_base < M) ? row_base : (M - 16);
  const int row = rb + l16;
  const float invc = 1.0f / fmaxf(Cnt[row], 1.0f);

  v8f acc[8] = {};
  for (int kb = 0; kb < 2; kb++) {
    if (kb) __syncthreads();                             // sW reads of chunk kb-1 done
    for (int i = tid; i < C_DIM * C_DIM / 4; i += 256)
      async_copy_b128(Wu + (size_t)kb * C_DIM * C_DIM + 4 * i, sW + 4 * i);
    wait_async0();
    __syncthreads();

    const float* Asrc = (kb == 0) ? Xd : Acc;
    const float scale = (kb == 0) ? 1.0f : invc;
    const float* Arow = Asrc + (size_t)row * C_DIM + 2 * half;
    v2f a_cur = *(const v2f*)(Arow);
    a_cur[0] *= scale; a_cur[1] *= scale;
    for (int k0 = 0; k0 < C_DIM; k0 += 4) {
      const int kn = (k0 + 4) & (C_DIM - 1);
      v2f a_nxt = *(const v2f*)(Arow + kn);
      a_nxt[0] *= scale; a_nxt[1] *= scale;
      const int ks = k0 + 2 * half;
      #pragma unroll
      for (int nt = 0; nt < 8; nt++) {
        const int n = nt * 16 + l16;
        v2f b;
        b[0] = sW[ks * C_DIM + n];
        b[1] = sW[(ks + 1) * C_DIM + n];
        acc[nt] = __builtin_amdgcn_wmma_f32_16x16x4_f32(false, a_cur, false, b,
                                                        (short)0, acc[nt], false, false);
      }
      a_cur = a_nxt;
    }
  }
  if (row_base >= M) return;
  #pragma unroll
  for (int nt = 0; nt < 8; nt++) {
    const int n = nt * 16 + l16;
    const float bv = bias[n];
    #pragma unroll
    for (int r = 0; r < 8; r++) {
      float z = acc[nt][r] + bv;
      float y = fmaxf(z, 0.0f);                           // relu
      if (z != z) y = 0.0f;                               // nan -> 0
      else if (__builtin_isinf(y)) y = 10.0f;             // +inf -> 10
      Y[(size_t)(row_base + r + 8 * half) * C_DIM + n] = y;
    }
  }
}

// ------------- per-edge: msg = LN(x_src[src] - P[dst]); atomic scatter to Acc[dst]
// one wave per edge, float4 per lane (32*4 = 128 channels)

__global__ __launch_bounds__(256) void edge_msg_kernel(const float* __restrict__ Xsrc,
                                                       const float* __restrict__ P,
                                                       const int* __restrict__ esrc,
                                                       const int* __restrict__ edst,
                                                       const float* __restrict__ g,
                                                       const float* __restrict__ be,
                                                       float* __restrict__ Acc, int ne) {
  const int wave = threadIdx.x >> 5, lane = threadIdx.x & 31;
  const int e = blockIdx.x * 8 + wave;
  if (e >= ne) return;
  const int s = esrc[e], d = edst[e];
  float4 xj = ((const float4*)(Xsrc + (size_t)s * C_DIM))[lane];
  float4 pr = ((const float4*)(P + (size_t)d * C_DIM))[lane];
  float4 r;
  r.x = xj.x - pr.x; r.y = xj.y - pr.y; r.z = xj.z - pr.z; r.w = xj.w - pr.w;
  float sum = wave_sum32(r.x + r.y + r.z + r.w);
  float sq  = wave_sum32(r.x * r.x + r.y * r.y + r.z * r.z + r.w * r.w);
  const float mean = sum * (1.0f / C_DIM);
  const float var = sq * (1.0f / C_DIM) - mean * mean;
  const float inv = rsqrtf(var + EPS_LN);
  float4 gg = ((const float4*)g)[lane];
  float4 bb = ((const float4*)be)[lane];
  float* dp = Acc + (size_t)d * C_DIM + lane * 4;
  atomicAdd(dp + 0, (r.x - mean) * inv * gg.x + bb.x);
  atomicAdd(dp + 1, (r.y - mean) * inv * gg.y + bb.y);
  atomicAdd(dp + 2, (r.z - mean) * inv * gg.z + bb.z);
  atomicAdd(dp + 3, (r.w - mean) * inv * gg.w + bb.w);
}

// ----------------------------- node LayerNorm + ReLU, one wave per row

__global__ __launch_bounds__(256) void ln_relu_kernel(const float* __restrict__ X,
                                                      const float* __restrict__ g,
                                                      const float* __restrict__ b,
                                                      float* __restrict__ Y, int M) {
  const int wave = threadIdx.x >> 5, lane = threadIdx.x & 31;
  const int row = blockIdx.x * 8 + wave;
  if (row >= M) return;
  float4 x = ((const float4*)(X + (size_t)row * C_DIM))[lane];
  float sum = wave_sum32(x.x + x.y + x.z + x.w);
  float sq  = wave_sum32(x.x * x.x + x.y * x.y + x.z * x.z + x.w * x.w);
  const float mean = sum * (1.0f / C_DIM);
  const float var = sq * (1.0f / C_DIM) - mean * mean;
  const float inv = rsqrtf(var + EPS_LN);
  float4 gg = ((const float4*)g)[lane];
  float4 bb = ((const float4*)b)[lane];
  float4 y;
  y.x = fmaxf((x.x - mean) * inv * gg.x + bb.x, 0.0f);
  y.y = fmaxf((x.y - mean) * inv * gg.y + bb.y, 0.0f);
  y.z = fmaxf((x.z - mean) * inv * gg.z + bb.z, 0.0f);
  y.w = fmaxf((x.w - mean) * inv * gg.w + bb.w, 0.0f);
  ((float4*)(Y + (size_t)row * C_DIM))[lane] = y;
}

// ----------------------------- head: out[i] = x_a[i] . hw + hb, one wave per row

__global__ __launch_bounds__(256) void head_kernel(const float* __restrict__ Xa,
                                                   const float* __restrict__ hw,
                                                   const float* __restrict__ hb,
                                                   float* __restrict__ out, int n) {
  const int wave = threadIdx.x >> 5, lane = threadIdx.x & 31;
  const int row = blockIdx.x * 8 + wave;
  if (row >= n) return;
  float4 x = ((const float4*)(Xa + (size_t)row * C_DIM))[lane];
  float4 w = ((const float4*)hw)[lane];
  float s = wave_sum32(x.x * w.x + x.y * w.y + x.z * w.z + x.w * w.w);
  if (lane == 0) out[row] = s + hb[0];
}

// ---------------------------------------------------------------- host side

static void run_conv(const float* xsrc, const float* xdst, const int* edge,
                     const float* w1, const float* b1, const float* w2, const float* b2,
                     const float* g, const float* be, const float* uw, const float* ub,
                     const float* cnt, float* H, float* P, float* out, int Mdst,
                     hipStream_t stream) {
  const dim3 blk(256);
  const int gG = (Mdst + 127) / 128;
  // pred MLP evaluated per-node (row-wise function => identical to per-edge)
  gemm128_kernel<<<gG, blk, 0, stream>>>(xdst, w1, b1, H, Mdst, 1);
  gemm128_kernel<<<gG, blk, 0, stream>>>(H, w2, b2, P, Mdst, 0);
  // H is free now: reuse as scatter accumulator
  const int nAcc = Mdst * C_DIM;
  zero_kernel<<<(nAcc + 255) / 256, blk, 0, stream>>>(H, nAcc);
  edge_msg_kernel<<<(E_N + 7) / 8, blk, 0, stream>>>(xsrc, P, edge, edge + E_N, g, be, H, E_N);
  update_gemm_kernel<<<gG, blk, 0, stream>>>(xdst, H, cnt, uw, ub, out, Mdst);
}

extern "C" void kernel_launch(void* const* d_in, const int* in_sizes, int n_in,
                              void* d_out, int out_size, void* d_ws, size_t ws_size,
                              hipStream_t stream) {
  const float* xin_a = (const float*)d_in[0];
  const float* xin_b = (const float*)d_in[1];
  const int* e_ab = (const int*)d_in[2];
  const int* e_ba = (const int*)d_in[3];
  const float* pw1 = (const float*)d_in[4];
  const float* pb1 = (const float*)d_in[5];
  const float* pw2 = (const float*)d_in[6];
  const float* pb2 = (const float*)d_in[7];
  const float* mg  = (const float*)d_in[8];
  const float* mbe = (const float*)d_in[9];
  const float* uw  = (const float*)d_in[10];
  const float* ub  = (const float*)d_in[11];
  const float* ng  = (const float*)d_in[12];
  const float* nb  = (const float*)d_in[13];
  const float* hw  = (const float*)d_in[14];
  const float* hb  = (const float*)d_in[15];
  (void)in_sizes; (void)n_in; (void)ws_size;

  float* ws = (float*)d_ws;
  const size_t NC = (size_t)NA_N * C_DIM;
  float* xa   = ws;            // 30000x128
  float* xb   = ws + NC;
  float* H    = ws + 2 * NC;   // hidden / accumulator (reused)
  float* P    = ws + 3 * NC;   // per-node pred
  float* nwA  = ws + 4 * NC;   // new a features (pre-LN)
  float* nwB  = ws + 5 * NC;   // new b features (pre-LN)
  float* cntA = ws + 6 * NC;   // in-degree of a (from edge_ba)
  float* cntB = cntA + NA_N;   // in-degree of b (from edge_ab)

  const dim3 blk(256);
  const int nNC = (int)NC;
  sanitize_in_kernel<<<(nNC + 255) / 256, blk, 0, stream>>>(xin_a, xa, nNC);
  sanitize_in_kernel<<<(nNC + 255) / 256, blk, 0, stream>>>(xin_b, xb, nNC);

  // edge lists are layer-invariant: count in-degrees once
  zero_kernel<<<(NA_N + 255) / 256, blk, 0, stream>>>(cntA, NA_N);
  zero_kernel<<<(NB_N + 255) / 256, blk, 0, stream>>>(cntB, NB_N);
  count_kernel<<<(E_N + 255) / 256, blk, 0, stream>>>(e_ab + E_N, cntB, E_N);
  count_kernel<<<(E_N + 255) / 256, blk, 0, stream>>>(e_ba + E_N, cntA, E_N);

  const int CC = C_DIM * C_DIM;
  for (int l = 0; l < L_NUM; l++) {
    const int i0 = l * 2 + 0, i1 = l * 2 + 1;
    // edge type 0: a -> b (dst = b), uses OLD xa, xb
    run_conv(xa, xb, e_ab,
             pw1 + (size_t)i0 * CC, pb1 + (size_t)i0 * C_DIM,
             pw2 + (size_t)i0 * CC, pb2 + (size_t)i0 * C_DIM,
             mg + (size_t)i0 * C_DIM, mbe + (size_t)i0 * C_DIM,
             uw + (size_t)i0 * 2 * CC, ub + (size_t)i0 * C_DIM,
             cntB, H, P, nwB, NB_N, stream);
    // edge type 1: b -> a (dst = a), uses OLD xa, xb
    run_conv(xb, xa, e_ba,
             pw1 + (size_t)i1 * CC, pb1 + (size_t)i1 * C_DIM,
             pw2 + (size_t)i1 * CC, pb2 + (size_t)i1 * C_DIM,
             mg + (size_t)i1 * C_DIM, mbe + (size_t)i1 * C_DIM,
             uw + (size_t)i1 * 2 * CC, ub + (size_t)i1 * C_DIM,
             cntA, H, P, nwA, NA_N, stream);
    // node LayerNorm + ReLU -> next-layer features
    ln_relu_kernel<<<(NA_N + 7) / 8, blk, 0, stream>>>(nwA, ng + (size_t)i0 * C_DIM,
                                                       nb + (size_t)i0 * C_DIM, xa, NA_N);
    ln_relu_kernel<<<(NB_N + 7) / 8, blk, 0, stream>>>(nwB, ng + (size_t)i1 * C_DIM,
                                                       nb + (size_t)i1 * C_DIM, xb, NB_N);
  }

  head_kernel<<<(out_size + 7) / 8, blk, 0, stream>>>(xa, hw, hb, (float*)d_out, out_size);
}